// AEGMM_LOSS_69355131896569
// MI455X (gfx1250) — compile-verified
//
#include <hip/hip_runtime.h>
#include <hip/hip_bf16.h>
#include <math.h>

typedef float v2f __attribute__((ext_vector_type(2)));
typedef float v4f __attribute__((ext_vector_type(4)));
typedef float v8f __attribute__((ext_vector_type(8)));

// ---------------- workspace layout (floats) ----------------
#define WS_GS    4      // gamma_sum[4]                       (atomic accum)
#define WS_MOM   8      // moments[2][16][16] from WMMA C     (atomic accum)
#define WS_PHI   520    // phi[4]
#define WS_ISD   524    // 1/sqrt(det_cov)[4]
#define WS_MU    528    // mu[4][8]
#define WS_CINV  560    // cov_inverse[4][8][8]
#define WS_CD    816    // cov_diag scalar
#define WS_MSEP  1024   // per-block MSE partials (<=2048)
#define WS_EP    4096   // per-block energy partials (<=2048)

#define EPS32    1.1920929e-07f
#define TWO_PI_F 6.2831853071795864769f

__device__ __forceinline__ float wave_reduce(float v) {
#pragma unroll
  for (int off = 16; off; off >>= 1) v += __shfl_xor(v, off, 32);
  return v;
}

__device__ float block_reduce_sum(float v) {
  __shared__ float sw[32];
  const int lane = threadIdx.x & 31;
  const int w = threadIdx.x >> 5;
  v = wave_reduce(v);
  if (lane == 0) sw[w] = v;
  __syncthreads();
  const int nw = (blockDim.x + 31) >> 5;
  v = (threadIdx.x < nw) ? sw[threadIdx.x] : 0.0f;
  if (w == 0) v = wave_reduce(v);
  __syncthreads();
  return v; // valid on thread 0
}

// ---------------- K1: zero accumulators ----------------
__global__ void init_ws_kernel(float* __restrict__ ws) {
  for (int i = threadIdx.x; i < 520; i += blockDim.x) ws[i] = 0.0f;
}

// ---------------- K2: MSE over 256 MB (non-temporal streaming) ----------------
__global__ void mse_kernel(const float* __restrict__ x, const float* __restrict__ xh,
                           float* __restrict__ partials, int n4) {
  const v4f* a4 = (const v4f*)x;
  const v4f* b4 = (const v4f*)xh;
  size_t i = (size_t)blockIdx.x * blockDim.x + threadIdx.x;
  const size_t stride = (size_t)gridDim.x * blockDim.x;
  float acc = 0.0f;
  for (; i < (size_t)n4; i += stride) {
    __builtin_prefetch(&a4[i + stride], 0, 0);          // global_prefetch_b8
    v4f a = __builtin_nontemporal_load(&a4[i]);         // TH=NT: don't pollute L2
    v4f b = __builtin_nontemporal_load(&b4[i]);
    float dx = b.x - a.x, dy = b.y - a.y, dz = b.z - a.z, dw = b.w - a.w;
    acc = fmaf(dx, dx, acc); acc = fmaf(dy, dy, acc);
    acc = fmaf(dz, dz, acc); acc = fmaf(dw, dw, acc);
  }
  float bs = block_reduce_sum(acc);
  if (threadIdx.x == 0) partials[blockIdx.x] = bs;
}

// ---------------- K3: GMM moments via V_WMMA_F32_16X16X4_F32 ----------------
// Per wave-step: 4 samples. A(16x4): rows = (component kk in half)*8 + dim d,
// value = gamma[n_j, k]*z[n_j, d].  B(4x16): cols 0..7 = z[n_j, e], col 8 = 1.
// C(16x16) accumulates: cols 0..7 -> Sigma gamma z z^T (2nd moment),
// col 8 -> Sigma gamma z (mu numerator). c0 covers k={0,1}, c1 covers k={2,3}.
__global__ void moments_wmma_kernel(const float* __restrict__ z,
                                    const float* __restrict__ gamma,
                                    float* __restrict__ ws, int N) {
  const int lane   = threadIdx.x & 31;
  const int wid    = threadIdx.x >> 5;
  const int wpb    = blockDim.x >> 5;
  const int gwave  = blockIdx.x * wpb + wid;
  const int nwaves = gridDim.x * wpb;
  const int nchunks = N >> 2;

  v8f c0 = {}; v8f c1 = {};
  float gs = 0.0f;

  const int j0 = (lane >> 4) * 2;  // sample pair handled by this half-wave
  const int m  = lane & 15;        // A row
  const int kk = m >> 3;           // component within half {0,1}
  const int d  = m & 7;            // z dimension
  const int e  = lane & 15;        // B/C column

  for (int ch = gwave; ch < nchunks; ch += nwaves) {
    const float zv = z[(size_t)ch * 32 + lane];                         // 4 samples x 8 dims, coalesced
    const float gv = (lane < 16) ? gamma[(size_t)ch * 16 + lane] : 0.f; // 4 samples x 4 comps

    // B operand (shared by both WMMAs)
    float bz0 = __shfl(zv, (j0 + 0) * 8 + (e & 7), 32);
    float bz1 = __shfl(zv, (j0 + 1) * 8 + (e & 7), 32);
    v2f b;
    b.x = (e < 8) ? bz0 : ((e == 8) ? 1.0f : 0.0f);
    b.y = (e < 8) ? bz1 : ((e == 8) ? 1.0f : 0.0f);

    // A operands for components {kk} and {2+kk}
    float zz0 = __shfl(zv, (j0 + 0) * 8 + d, 32);
    float zz1 = __shfl(zv, (j0 + 1) * 8 + d, 32);
    float g00 = __shfl(gv, (j0 + 0) * 4 + kk, 32);
    float g01 = __shfl(gv, (j0 + 1) * 4 + kk, 32);
    float g10 = __shfl(gv, (j0 + 0) * 4 + 2 + kk, 32);
    float g11 = __shfl(gv, (j0 + 1) * 4 + 2 + kk, 32);
    v2f a0, a1;
    a0.x = g00 * zz0; a0.y = g01 * zz1;
    a1.x = g10 * zz0; a1.y = g11 * zz1;

#if defined(__has_builtin) && __has_builtin(__builtin_amdgcn_wmma_f32_16x16x4_f32)
    c0 = __builtin_amdgcn_wmma_f32_16x16x4_f32(false, a0, false, b, (short)0, c0, false, false);
    c1 = __builtin_amdgcn_wmma_f32_16x16x4_f32(false, a1, false, b, (short)0, c1, false, false);
#else
    { // fallback: same C layout, manual MACs
      const int col = lane & 15;
      const int rhi = (lane >> 4) * 8;
#pragma unroll
      for (int r = 0; r < 8; ++r) {
        const int row = r + rhi; const int kk2 = row >> 3; const int d2 = row & 7;
#pragma unroll
        for (int j = 0; j < 4; ++j) {
          float zjd  = __shfl(zv, j * 8 + d2, 32);
          float bcol = __shfl(zv, j * 8 + (col & 7), 32);
          float bj   = (col < 8) ? bcol : ((col == 8) ? 1.0f : 0.0f);
          float gj0  = __shfl(gv, j * 4 + kk2, 32);
          float gj1  = __shfl(gv, j * 4 + 2 + kk2, 32);
          c0[r] = fmaf(gj0 * zjd, bj, c0[r]);
          c1[r] = fmaf(gj1 * zjd, bj, c1[r]);
        }
      }
    }
#endif
    gs += gv;
  }

  // block-level reduction in LDS (ds_add_f32), then few global atomics
  __shared__ float sm[516];
  for (int i = threadIdx.x; i < 516; i += blockDim.x) sm[i] = 0.0f;
  __syncthreads();
  const int col = lane & 15;
  const int rhi = (lane >> 4) * 8;   // C layout: VGPR r -> row r (lanes 0-15) / r+8 (lanes 16-31)
#pragma unroll
  for (int r = 0; r < 8; ++r) {
    atomicAdd(&sm[0 * 256 + (r + rhi) * 16 + col], c0[r]);
    atomicAdd(&sm[1 * 256 + (r + rhi) * 16 + col], c1[r]);
  }
  if (lane < 16) atomicAdd(&sm[512 + (lane & 3)], gs);
  __syncthreads();
  for (int i = threadIdx.x; i < 512; i += blockDim.x) atomicAdd(&ws[WS_MOM + i], sm[i]);
  if (threadIdx.x < 4) atomicAdd(&ws[WS_GS + threadIdx.x], sm[512 + threadIdx.x]);
}

// ---------------- K4: phi/mu/cov, 8x8 inverse, Cholesky det (1 block) ----------
__global__ void finalize_params_kernel(float* __restrict__ ws, int N) {
  __shared__ float cov[4][8][8];
  __shared__ float aug[4][8][16];
  __shared__ float mu_s[4][8];
  __shared__ float gsum_s[4];
  const int t = threadIdx.x;
  if (t < 4) gsum_s[t] = ws[WS_GS + t];
  __syncthreads();
  if (t < 32) { // mu = S1 / (gsum + eps)
    int k = t >> 3, dd = t & 7, h = k >> 1, kk = k & 1;
    float S1 = ws[WS_MOM + h * 256 + (kk * 8 + dd) * 16 + 8];
    mu_s[k][dd] = S1 / (gsum_s[k] + EPS32);
  }
  __syncthreads();
  if (t < 256) { // centered covariance + eps*I, build augmented system
    int k = t >> 6, r = t & 63, dd = r >> 3, ee = r & 7, h = k >> 1, kk = k & 1;
    float S2  = ws[WS_MOM + h * 256 + (kk * 8 + dd) * 16 + ee];
    float S1d = ws[WS_MOM + h * 256 + (kk * 8 + dd) * 16 + 8];
    float S1e = ws[WS_MOM + h * 256 + (kk * 8 + ee) * 16 + 8];
    float gsm = gsum_s[k];
    float mud = mu_s[k][dd], mue = mu_s[k][ee];
    float num = S2 - mud * S1e - mue * S1d + gsm * mud * mue;
    float c = num / (gsm + EPS32);
    if (dd == ee) c += 1e-12f;
    cov[k][dd][ee] = c;
    aug[k][dd][ee] = c;
    aug[k][dd][8 + ee] = (dd == ee) ? 1.0f : 0.0f;
  }
  __syncthreads();
  if (t == 0) { // cov_diag = sum 1/diag
    float cd = 0.0f;
    for (int k = 0; k < 4; ++k)
      for (int dd = 0; dd < 8; ++dd) cd += 1.0f / cov[k][dd][dd];
    ws[WS_CD] = cd;
  }
  if (t < 4) { // per-component: Gauss-Jordan inverse + Cholesky det of 2*pi*cov
    const int k = t;
    for (int p = 0; p < 8; ++p) {
      float inv = 1.0f / aug[k][p][p];
      for (int c2 = 0; c2 < 16; ++c2) aug[k][p][c2] *= inv;
      for (int r = 0; r < 8; ++r) if (r != p) {
        float f = aug[k][r][p];
        for (int c2 = 0; c2 < 16; ++c2) aug[k][r][c2] = fmaf(-f, aug[k][p][c2], aug[k][r][c2]);
      }
    }
    float L[8][8];
    for (int r = 0; r < 8; ++r)
      for (int c2 = 0; c2 < 8; ++c2) L[r][c2] = TWO_PI_F * cov[k][r][c2];
    float det = 1.0f;
    for (int c2 = 0; c2 < 8; ++c2) {
      float s = L[c2][c2];
      for (int j = 0; j < c2; ++j) s = fmaf(-L[c2][j], L[c2][j], s);
      float l = sqrtf(s);
      det *= l;
      float invl = 1.0f / l;
      L[c2][c2] = l;
      for (int r = c2 + 1; r < 8; ++r) {
        float s2 = L[r][c2];
        for (int j = 0; j < c2; ++j) s2 = fmaf(-L[r][j], L[c2][j], s2);
        L[r][c2] = s2 * invl;
      }
    }
    ws[WS_PHI + k] = gsum_s[k] / ((float)N + EPS32);
    ws[WS_ISD + k] = 1.0f / sqrtf(det);  // reference: E /= sqrt(prod diag chol(2*pi*cov))
  }
  __syncthreads();
  if (t < 32) { int k = t >> 3, dd = t & 7; ws[WS_MU + k * 8 + dd] = mu_s[k][dd]; }
  if (t < 256) { int k = t >> 6, r = t & 63, dd = r >> 3, ee = r & 7;
                 ws[WS_CINV + k * 64 + r] = aug[k][dd][8 + ee]; }
}

// ---------------- K5: per-sample energy ----------------
__global__ void energy_kernel(const float* __restrict__ z, const float* __restrict__ ws,
                              float* __restrict__ partials, int N) {
  __shared__ float P[296]; // phi[4] | isd[4] | mu[32] | cinv[256]
  for (int i = threadIdx.x; i < 296; i += blockDim.x) P[i] = ws[WS_PHI + i];
  __syncthreads();
  const int n = blockIdx.x * blockDim.x + threadIdx.x;
  float E = 0.0f;
  if (n < N) {
    const v4f* z4 = (const v4f*)(z + (size_t)n * 8);
    v4f za = z4[0], zb = z4[1];
    float zr[8] = { za.x, za.y, za.z, za.w, zb.x, zb.y, zb.z, zb.w };
    float s = 0.0f;
#pragma unroll
    for (int k = 0; k < 4; ++k) {
      float diff[8];
#pragma unroll
      for (int dd = 0; dd < 8; ++dd) diff[dd] = zr[dd] - P[8 + k * 8 + dd];
      float q = 0.0f;
#pragma unroll
      for (int dd = 0; dd < 8; ++dd) {
        float tt = 0.0f;
#pragma unroll
        for (int ee = 0; ee < 8; ++ee) tt = fmaf(P[40 + k * 64 + dd * 8 + ee], diff[ee], tt);
        q = fmaf(diff[dd], tt, q);
      }
      s = fmaf(P[k] * P[4 + k], expf(-0.5f * q), s);
    }
    E = -logf(s + 1e-12f);
  }
  float bs = block_reduce_sum(E);
  if (threadIdx.x == 0) partials[blockIdx.x] = bs;
}

// ---------------- K6: combine ----------------
__global__ void final_kernel(const float* __restrict__ ws, float* __restrict__ out,
                             float inv_nx, float inv_n, int mse_blocks, int e_blocks) {
  float m = 0.0f, e = 0.0f;
  for (int i = threadIdx.x; i < mse_blocks; i += blockDim.x) m += ws[WS_MSEP + i];
  for (int i = threadIdx.x; i < e_blocks; i += blockDim.x) e += ws[WS_EP + i];
  float ms = block_reduce_sum(m);
  float es = block_reduce_sum(e);
  if (threadIdx.x == 0)
    out[0] = ms * inv_nx + 0.1f * (es * inv_n) + 0.005f * ws[WS_CD];
}

extern "C" void kernel_launch(void* const* d_in, const int* in_sizes, int n_in,
                              void* d_out, int out_size, void* d_ws, size_t ws_size,
                              hipStream_t stream) {
  const float* x  = (const float*)d_in[0];
  const float* xh = (const float*)d_in[1];
  const float* z  = (const float*)d_in[2];
  const float* g  = (const float*)d_in[3];
  float* ws  = (float*)d_ws;
  float* out = (float*)d_out;

  const int nx = in_sizes[0];       // B*C*T*F = 33,554,432
  const int N  = in_sizes[3] / 4;   // B*C*T   = 262,144 samples (K=4)
  const int n4 = nx / 4;

  const int MSE_BLOCKS = 2048;
  const int E_BLOCKS   = (N + 255) / 256;

  init_ws_kernel<<<1, 256, 0, stream>>>(ws);
  mse_kernel<<<MSE_BLOCKS, 256, 0, stream>>>(x, xh, ws + WS_MSEP, n4);
  moments_wmma_kernel<<<256, 256, 0, stream>>>(z, g, ws, N);
  finalize_params_kernel<<<1, 256, 0, stream>>>(ws, N);
  energy_kernel<<<E_BLOCKS, 256, 0, stream>>>(z, ws, ws + WS_EP, N);
  final_kernel<<<1, 256, 0, stream>>>(ws, out, 1.0f / (float)nx, 1.0f / (float)N,
                                      MSE_BLOCKS, E_BLOCKS);
}